// SSIMLoss_24567212933326
// MI455X (gfx1250) — compile-verified
//
#include <hip/hip_runtime.h>

// SSIM loss, fused: 5 depthwise 11x11 Gaussian convs (separable) done as
// V_WMMA_F32_16X16X4_F32 banded matmuls + fused SSIM map + deterministic
// reduction. Memory floor ~4.1us @23.3TB/s; WMMA keeps fp32 conv compute
// near that floor (1024 MAC/instr vs 32 for v_fma_f32).

typedef __attribute__((ext_vector_type(2))) float v2f;
typedef __attribute__((ext_vector_type(8))) float v8f;

#define IMG_H 512
#define IMG_W 512
#define N_PLANES 48                      // B*C = 16*3
#define REG_H 32
#define REG_W 64
#define REGS_Y (IMG_H / REG_H)           // 16
#define REGS_X (IMG_W / REG_W)           // 8
#define NBLOCKS (N_PLANES * REGS_Y * REGS_X)   // 6144
#define HB_STRIDE 19                     // Hbuf row stride (conflict-free)

__global__ __launch_bounds__(256)
void ssim_tiles(const float* __restrict__ pred,
                const float* __restrict__ tgt,
                const float* __restrict__ win,
                float* __restrict__ partials)
{
  __shared__ float GT[48];               // g[] with zero guard bands
  __shared__ float Sp[42 * 76];          // staged pred region (+5 halo)
  __shared__ float St[42 * 76];          // staged target region
  __shared__ float Hb[8][28 * HB_STRIDE];// per-wave horizontal-pass spill
  __shared__ float red[256];

  const int tid = threadIdx.x;
  const int bi  = blockIdx.x;
  const int plane = bi >> 7;             // / 128 regions per plane
  const int rem   = bi & 127;
  const int regY  = rem >> 3;
  const int regX  = rem & 7;
  const int row0  = regY * REG_H - 5;    // staged row r -> image row row0+r
  const int col0  = regX * REG_W - 5;
  const size_t pOff = (size_t)plane * (IMG_H * IMG_W);

  // 1D gaussian from the 2D window: g[i] = w2d[i][5] / sqrt(w2d[5][5]).
  // GT[16+d] = g[d] for d in [0,10], zero elsewhere -> branch-free band.
  if (tid < 48) {
    int d = tid - 16;
    float v = 0.0f;
    if (d >= 0 && d <= 10) v = win[d * 11 + 5] / sqrtf(win[60]);
    GT[tid] = v;
  }

  // Cooperative stage of 42x76 (cols 74/75 are slack hit only by zero
  // weights; keep them finite -> bounds-check, else 0).
  for (int idx = tid; idx < 42 * 76; idx += 256) {
    int r = idx / 76, c = idx - r * 76;
    int gr = row0 + r, gc = col0 + c;
    bool ok = (gr >= 0) & (gr < IMG_H) & (gc >= 0) & (gc < IMG_W);
    size_t g = pOff + (size_t)gr * IMG_W + (size_t)gc;
    Sp[idx] = ok ? pred[g] : 0.0f;
    St[idx] = ok ? tgt[g]  : 0.0f;
  }
  __syncthreads();

  const int wave = tid >> 5;
  const int lane = tid & 31;
  const int hi   = lane >> 4;            // half-wave select (K block)
  const int l16  = lane & 15;            // M (A) / N (B) index
  const int tr   = wave >> 2;            // tile row 0..1
  const int tc   = wave & 3;             // tile col 0..3
  float* hb = Hb[wave];

  // Zero Hbuf guard rows 26..27 (read by vertical chunk 6 with zero weight;
  // must be finite, WMMA propagates NaN even through 0*x).
  for (int idx = 26 * HB_STRIDE + lane; idx < 28 * HB_STRIDE; idx += 32)
    hb[idx] = 0.0f;

  // Banded weight operands. Wv = Wh^T and the f32 A/B lane layouts are
  // symmetric (lane&15 = M or N, K = 4*chunk + 2*hi + vgpr), so ONE register
  // pair serves as horizontal-B and vertical-A:  w[k][q] = g[(4k+2hi+q)-l16].
  v2f wv[7];
#pragma unroll
  for (int k = 0; k < 7; ++k) {
    int u0 = 4 * k + 2 * hi;
    v2f w;
    w[0] = GT[16 + u0     - l16];
    w[1] = GT[16 + u0 + 1 - l16];
    wv[k] = w;
  }

  const int rowA = 16 * tr + l16;        // A-operand row in staged coords
  const int colB = 16 * tc + 2 * hi;     // A-operand col base

  v8f E[5];
  const v8f vzero = {0, 0, 0, 0, 0, 0, 0, 0};
  const int hoff[2] = {0, 10};           // two 16-row tiles cover 26 rows

#pragma unroll
  for (int pl = 0; pl < 5; ++pl) {
    // ---- horizontal pass: Hp[j][c] = sum_u S[j][u] * g[u-c], j=0..25 ----
    v8f accH0 = vzero, accH1 = vzero;
#pragma unroll
    for (int h = 0; h < 2; ++h) {
      const int roff = hoff[h];
#pragma unroll
      for (int k = 0; k < 7; ++k) {
        const int o = (rowA + roff) * 76 + colB + 4 * k;  // 8B aligned
        float p0 = Sp[o], p1 = Sp[o + 1];
        float t0 = St[o], t1 = St[o + 1];
        v2f a;
        if      (pl == 0) { a[0] = p0;      a[1] = p1;      }
        else if (pl == 1) { a[0] = t0;      a[1] = t1;      }
        else if (pl == 2) { a[0] = p0 * p0; a[1] = p1 * p1; }
        else if (pl == 3) { a[0] = t0 * t0; a[1] = t1 * t1; }
        else              { a[0] = p0 * t0; a[1] = p1 * t1; }
        if (h == 0)
          accH0 = __builtin_amdgcn_wmma_f32_16x16x4_f32(
              false, a, false, wv[k], (short)0, accH0, false, false);
        else
          accH1 = __builtin_amdgcn_wmma_f32_16x16x4_f32(
              false, a, false, wv[k], (short)0, accH1, false, false);
      }
    }
    // ---- spill Hp rows 0..25 (C-layout -> row-major LDS) ----
#pragma unroll
    for (int i = 0; i < 8; ++i)
      hb[(i + 8 * hi) * HB_STRIDE + l16] = accH0[i];
#pragma unroll
    for (int i = 0; i < 8; ++i)
      hb[(i + 8 * hi + 10) * HB_STRIDE + l16] = accH1[i];

    // ---- vertical pass: out[m][n] = sum_j g[j-m] * Hp[j][n] ----
    v8f acc = vzero;
#pragma unroll
    for (int k = 0; k < 7; ++k) {
      const int j0 = 4 * k + 2 * hi;
      v2f b;
      b[0] = hb[ j0      * HB_STRIDE + l16];
      b[1] = hb[(j0 + 1) * HB_STRIDE + l16];
      acc = __builtin_amdgcn_wmma_f32_16x16x4_f32(
          false, wv[k], false, b, (short)0, acc, false, false);
    }
    E[pl] = acc;
  }

  // ---- fused SSIM map over this lane's 8 output pixels ----
  const float C1 = 0.01f * 0.01f, C2 = 0.03f * 0.03f;
  float part = 0.0f;
#pragma unroll
  for (int i = 0; i < 8; ++i) {
    float mu1 = E[0][i], mu2 = E[1][i];
    float s1  = E[2][i] - mu1 * mu1;
    float s2  = E[3][i] - mu2 * mu2;
    float s12 = E[4][i] - mu1 * mu2;
    float num = (2.0f * mu1 * mu2 + C1) * (2.0f * s12 + C2);
    float den = (mu1 * mu1 + mu2 * mu2 + C1) * (s1 + s2 + C2);
    part += num / den;
  }

  // ---- deterministic block reduction (fixed-order tree, no atomics) ----
  red[tid] = part;
  __syncthreads();
#pragma unroll
  for (int s = 128; s > 0; s >>= 1) {
    if (tid < s) red[tid] += red[tid + s];
    __syncthreads();
  }
  if (tid == 0) partials[bi] = red[0];
}

__global__ __launch_bounds__(256)
void ssim_finalize(const float* __restrict__ partials, float* __restrict__ out)
{
  __shared__ double rd[256];
  const int tid = threadIdx.x;
  double s = 0.0;
  for (int i = tid; i < NBLOCKS; i += 256) s += (double)partials[i];
  rd[tid] = s;
  __syncthreads();
  for (int st = 128; st > 0; st >>= 1) {
    if (tid < st) rd[tid] += rd[tid + st];
    __syncthreads();
  }
  if (tid == 0)
    out[0] = (float)(1.0 - rd[0] / (double)(16.0 * 3.0 * 512.0 * 512.0));
}

extern "C" void kernel_launch(void* const* d_in, const int* in_sizes, int n_in,
                              void* d_out, int out_size, void* d_ws, size_t ws_size,
                              hipStream_t stream) {
  (void)in_sizes; (void)n_in; (void)out_size; (void)ws_size;
  const float* pred = (const float*)d_in[0];
  const float* tgt  = (const float*)d_in[1];
  const float* win  = (const float*)d_in[2];
  float* partials   = (float*)d_ws;      // 6144 floats of scratch

  ssim_tiles<<<NBLOCKS, 256, 0, stream>>>(pred, tgt, win, partials);
  ssim_finalize<<<1, 256, 0, stream>>>(partials, (float*)d_out);
}